// StreamingAttention_80298708566042
// MI455X (gfx1250) — compile-verified
//
#include <hip/hip_runtime.h>

#define D_MODEL   1024
#define N_HEADS   16
#define HEAD_DIM  64
#define B_SZ      8
#define L_SZ      16
#define S_CACHE   8192
#define S_TOT     (S_CACHE + L_SZ)   /* 8208 */
#define N_CHUNKS  (S_TOT / 16)       /* 513  */
#define ROWS      (B_SZ * L_SZ)      /* 128  */
#define KPAD      68                 /* 64 + 4 pad floats per LDS row */

typedef float v2f __attribute__((ext_vector_type(2)));
typedef float v8f __attribute__((ext_vector_type(8)));

#define WMMA_F32(a, b, c) \
  __builtin_amdgcn_wmma_f32_16x16x4_f32(false, (a), false, (b), (short)0, (c), false, false)

// ---- CDNA5 async global->LDS copy (ASYNCcnt-tracked), with sync fallback ----
#if __has_builtin(__builtin_amdgcn_global_load_async_to_lds_b128)
typedef int async_v4i __attribute__((vector_size(16)));
typedef __attribute__((address_space(1))) async_v4i* async_gptr;
typedef __attribute__((address_space(3))) async_v4i* async_lptr;
#define ASYNC_COPY_B128(ldsp, gp)                                           \
  __builtin_amdgcn_global_load_async_to_lds_b128(                           \
      (async_gptr)(gp), (async_lptr)(ldsp), 0, 0)
#else
#define ASYNC_COPY_B128(ldsp, gp)                                           \
  do { *(float4*)(ldsp) = *(const float4*)(gp); } while (0)
#endif

#if __has_builtin(__builtin_amdgcn_s_wait_asynccnt)
#define WAIT_ASYNC(n) __builtin_amdgcn_s_wait_asynccnt(n)
#else
#define WAIT_ASYNC(n) asm volatile("s_wait_asynccnt %0" ::"i"(n) : "memory")
#endif

#define WAIT_DS0() asm volatile("s_wait_dscnt 0x0" ::: "memory")

// ---------------------------------------------------------------------------
// C = A(128 x K) * W(K x N) + bias, fp32 WMMA 16x16x4.
// Block = 128 threads = 4 waves; each wave computes one 16x16 tile.
// ---------------------------------------------------------------------------
__global__ void gemm_bias_wmma(const float* __restrict__ A,
                               const float* __restrict__ W,
                               const float* __restrict__ bias,
                               float* __restrict__ out,
                               int K, int N) {
  const int lane = threadIdx.x & 31;
  const int wave = threadIdx.x >> 5;
  const int half = lane >> 4;
  const int l16  = lane & 15;

  const int ntile = blockIdx.x * 4 + wave;
  const int mtile = blockIdx.y;
  const int arow  = mtile * 16 + l16;
  const int ncol  = ntile * 16 + l16;

  v8f acc = {};
  for (int kk = 0; kk < K; kk += 4) {
    const int kc = kk + 2 * half;
    v2f a = *(const v2f*)(A + (size_t)arow * K + kc);
    v2f b;
    b.x = W[(size_t)kc * N + ncol];
    b.y = W[(size_t)(kc + 1) * N + ncol];
    acc = WMMA_F32(a, b, acc);
  }

  const float bz = bias[ncol];
#pragma unroll
  for (int v = 0; v < 8; ++v) {
    const int r = mtile * 16 + v + 8 * half;
    out[(size_t)r * N + ncol] = acc[v] + bz;
  }
}

// ---------------------------------------------------------------------------
// RoPE on q/k slices of the raw qkv buffer; scatter into [B][H][L][D] layout.
// ---------------------------------------------------------------------------
__global__ void rope_scatter(const float* __restrict__ qkv,
                             const int* __restrict__ cidx,
                             float* __restrict__ qr,
                             float* __restrict__ kr,
                             float* __restrict__ vn) {
  const int bl = blockIdx.x;             // 0..127
  const int b  = bl / L_SZ;
  const int l  = bl % L_SZ;
  const float pos = (float)(cidx[0] + l);
  const float* row = qkv + (size_t)bl * (3 * D_MODEL);

  for (int idx = threadIdx.x; idx < D_MODEL; idx += blockDim.x) {
    const int h = idx >> 6;
    const int d = idx & 63;
    const int j = d & 31;
    const float inv = __powf(10000.0f, -(float)j / 32.0f);
    const float fr  = pos * inv;
    const float c = __cosf(fr);
    const float s = __sinf(fr);

    const int base = h * HEAD_DIM;
    const float qd = row[idx];
    const float kd = row[D_MODEL + idx];
    const float vd = row[2 * D_MODEL + idx];

    float qo, ko;
    if (d < 32) {
      const float q2 = row[base + d + 32];
      const float k2 = row[D_MODEL + base + d + 32];
      qo = qd * c - q2 * s;
      ko = kd * c - k2 * s;
    } else {
      const float q1 = row[base + d - 32];
      const float k1 = row[D_MODEL + base + d - 32];
      qo = qd * c + q1 * s;
      ko = kd * c + k1 * s;
    }
    const size_t o = (((size_t)b * N_HEADS + h) * L_SZ + l) * HEAD_DIM + d;
    qr[o] = qo;
    kr[o] = ko;
    vn[o] = vd;
  }
}

// ---------------------------------------------------------------------------
// Fused flash attention with async double-buffered K/V staging in LDS.
// One block per (b,h); 4 waves stripe the 513 S-chunks (16 wide).
// ---------------------------------------------------------------------------
__global__ void __launch_bounds__(128)
flash_attn(const float* __restrict__ qr,
           const float* __restrict__ kr,
           const float* __restrict__ vn,
           const float* __restrict__ cache_k,
           const float* __restrict__ cache_v,
           float* __restrict__ attn) {
  const int bh   = blockIdx.x;           // 0..127
  const int b    = bh / N_HEADS;
  const int h    = bh % N_HEADS;
  const int lane = threadIdx.x & 31;
  const int wave = threadIdx.x >> 5;
  const int half = lane >> 4;
  const int l16  = lane & 15;

  __shared__ float ktile[4][2][16][KPAD];        // per-wave double-buffered K
  __shared__ float vtile[4][2][16][KPAD];        // per-wave double-buffered V
  __shared__ float pbuf[4][16][17];              // P tile staging (padded)
  __shared__ float m_sh[4][16];                  // per-wave row max
  __shared__ float s_sh[4][16];                  // per-wave row sum
  __shared__ float acc_sh[4][16][HEAD_DIM];      // per-wave partial output

  // ---- preload Q tile as 16 A-fragments (K-steps of 4 over d=0..63) ----
  const float* qbase = qr + (((size_t)b * N_HEADS + h) * L_SZ) * HEAD_DIM;
  v2f qf[16];
#pragma unroll
  for (int t = 0; t < 16; ++t) {
    const int kc = 4 * t + 2 * half;
    qf[t] = *(const v2f*)(qbase + (size_t)l16 * HEAD_DIM + kc);
  }

  const size_t cstride = (size_t)N_HEADS * HEAD_DIM;  // 1024 floats/row
  const float* kc_base = cache_k + (size_t)b * S_CACHE * cstride + h * HEAD_DIM;
  const float* vc_base = cache_v + (size_t)b * S_CACHE * cstride + h * HEAD_DIM;
  const float* kn_base = kr + (((size_t)b * N_HEADS + h) * L_SZ) * HEAD_DIM;
  const float* vn_base = vn + (((size_t)b * N_HEADS + h) * L_SZ) * HEAD_DIM;

  // Issue 16 async b128 copies moving one 16x64 K chunk + V chunk into LDS.
  // Each op: 32 lanes x 16B = 2 rows of 256B.
  auto issue_chunk = [&](int c, int buf) {
    const float* kp;
    const float* vp;
    size_t kstride;
    if (c < S_CACHE / 16) {
      kp = kc_base + (size_t)c * 16 * cstride;
      vp = vc_base + (size_t)c * 16 * cstride;
      kstride = cstride;
    } else {
      kp = kn_base;
      vp = vn_base;
      kstride = HEAD_DIM;
    }
#pragma unroll
    for (int i = 0; i < 8; ++i) {
      const int row = 2 * i + half;
      const float* gk = kp + (size_t)row * kstride + l16 * 4;
      const float* gv = vp + (size_t)row * kstride + l16 * 4;
      ASYNC_COPY_B128(&ktile[wave][buf][row][l16 * 4], gk);
      ASYNC_COPY_B128(&vtile[wave][buf][row][l16 * 4], gv);
    }
  };

  float mrow[8], lrow[8];
#pragma unroll
  for (int v = 0; v < 8; ++v) { mrow[v] = -INFINITY; lrow[v] = 0.0f; }
  v8f oacc[4] = {};

  const float scale = 0.125f;  // 1/sqrt(HEAD_DIM)

  int buf = 0;
  issue_chunk(wave, 0);

  for (int c = wave; c < N_CHUNKS; c += 4) {
    const bool have_next = (c + 4) < N_CHUNKS;

    // deep prefetch into L2 for the chunk after next (wave-uniform branch)
    if (c + 8 < S_CACHE / 16) {
      const size_t nro = ((size_t)(c + 8) * 16 + l16) * cstride + half * 32;
      __builtin_prefetch((const char*)(kc_base + nro), 0, 0);
      __builtin_prefetch((const char*)(vc_base + nro), 0, 0);
    }

    // all LDS reads of the buffer we are about to overwrite have completed
    WAIT_DS0();
    if (have_next) {
      issue_chunk(c + 4, buf ^ 1);
      WAIT_ASYNC(16);    // current chunk resident; next chunk in flight
    } else {
      WAIT_ASYNC(0);
    }
    __builtin_amdgcn_wave_barrier();

    // ---- scores S(16x16) = Q(16x64) * K^T(64x16) ----
    v8f sc = {};
#pragma unroll
    for (int t = 0; t < 16; ++t) {
      const int kc2 = 4 * t + 2 * half;
      // b[v] = K[n=l16][kc2+v] from LDS
      v2f bf = *(const v2f*)(&ktile[wave][buf][l16][kc2]);
      sc = WMMA_F32(qf[t], bf, sc);
    }

    // ---- online softmax over this chunk ----
    float alpha[8];
#pragma unroll
    for (int v = 0; v < 8; ++v) {
      const float sv = sc[v] * scale;
      float rmax = sv;
#pragma unroll
      for (int off = 1; off < 16; off <<= 1)
        rmax = fmaxf(rmax, __shfl_xor(rmax, off, 32));
      const float mnew = fmaxf(mrow[v], rmax);
      const float p = __expf(sv - mnew);
      float rsum = p;
#pragma unroll
      for (int off = 1; off < 16; off <<= 1)
        rsum += __shfl_xor(rsum, off, 32);
      alpha[v] = __expf(mrow[v] - mnew);
      lrow[v]  = lrow[v] * alpha[v] + rsum;
      mrow[v]  = mnew;
      pbuf[wave][v + 8 * half][l16] = p;   // C-layout row/col
    }
    __builtin_amdgcn_wave_barrier();

    // rescale existing accumulators
#pragma unroll
    for (int t = 0; t < 4; ++t)
#pragma unroll
      for (int v = 0; v < 8; ++v)
        oacc[t][v] *= alpha[v];

    // ---- O(16x64) += P(16x16) * V(16x64) ----
#pragma unroll
    for (int ks = 0; ks < 4; ++ks) {
      const int kc2 = 4 * ks + 2 * half;
      v2f pf;
      pf.x = pbuf[wave][l16][kc2];
      pf.y = pbuf[wave][l16][kc2 + 1];
#pragma unroll
      for (int t = 0; t < 4; ++t) {
        v2f vf;
        vf.x = vtile[wave][buf][kc2][t * 16 + l16];
        vf.y = vtile[wave][buf][kc2 + 1][t * 16 + l16];
        oacc[t] = WMMA_F32(pf, vf, oacc[t]);
      }
    }
    __builtin_amdgcn_wave_barrier();
    buf ^= 1;
  }

  // ---- dump per-wave partials ----
#pragma unroll
  for (int v = 0; v < 8; ++v) {
    const int r = v + 8 * half;
    if (l16 == 0) {
      m_sh[wave][r] = mrow[v];
      s_sh[wave][r] = lrow[v];
    }
#pragma unroll
    for (int t = 0; t < 4; ++t)
      acc_sh[wave][r][t * 16 + l16] = oacc[t][v];
  }
  __syncthreads();

  // ---- merge 4 wave-partials (log-sum-exp) and write out ----
  const int tid   = threadIdx.x;
  const int r     = tid >> 3;          // query row 0..15
  const int dbase = (tid & 7) * 8;     // 8 consecutive d's
  float M = -INFINITY;
#pragma unroll
  for (int w = 0; w < 4; ++w) M = fmaxf(M, m_sh[w][r]);
  float wfac[4];
  float den = 0.0f;
#pragma unroll
  for (int w = 0; w < 4; ++w) {
    wfac[w] = __expf(m_sh[w][r] - M);
    den += wfac[w] * s_sh[w][r];
  }
  const float invden = 1.0f / den;
#pragma unroll
  for (int dd = 0; dd < 8; ++dd) {
    const int d = dbase + dd;
    float num = 0.0f;
#pragma unroll
    for (int w = 0; w < 4; ++w) num += wfac[w] * acc_sh[w][r][d];
    attn[((size_t)(b * L_SZ + r)) * D_MODEL + h * HEAD_DIM + d] = num * invden;
  }
}

// ---------------------------------------------------------------------------
extern "C" void kernel_launch(void* const* d_in, const int* in_sizes, int n_in,
                              void* d_out, int out_size, void* d_ws, size_t ws_size,
                              hipStream_t stream) {
  const float* x       = (const float*)d_in[0];
  const float* cache_k = (const float*)d_in[1];
  const float* cache_v = (const float*)d_in[2];
  const float* qkv_w   = (const float*)d_in[3];
  const float* qkv_b   = (const float*)d_in[4];
  const float* o_w     = (const float*)d_in[5];
  const float* o_b     = (const float*)d_in[6];
  const int*   cidx    = (const int*)d_in[7];
  float* out = (float*)d_out;

  float* ws      = (float*)d_ws;
  float* qkv_raw = ws;                                   // 128*3072
  float* qr      = qkv_raw + (size_t)ROWS * 3 * D_MODEL; // 131072 each
  float* kr      = qr + (size_t)B_SZ * N_HEADS * L_SZ * HEAD_DIM;
  float* vn      = kr + (size_t)B_SZ * N_HEADS * L_SZ * HEAD_DIM;
  float* attn    = vn + (size_t)B_SZ * N_HEADS * L_SZ * HEAD_DIM;

  // 1) QKV projection (fp32 WMMA)
  gemm_bias_wmma<<<dim3((3 * D_MODEL) / 64, ROWS / 16), 128, 0, stream>>>(
      x, qkv_w, qkv_b, qkv_raw, D_MODEL, 3 * D_MODEL);

  // 2) RoPE + scatter
  rope_scatter<<<dim3(ROWS), 256, 0, stream>>>(qkv_raw, cidx, qr, kr, vn);

  // 3) Fused flash attention over cache + new tokens (async LDS staging)
  flash_attn<<<dim3(B_SZ * N_HEADS), 128, 0, stream>>>(
      qr, kr, vn, cache_k, cache_v, attn);

  // 4) Output projection (fp32 WMMA)
  gemm_bias_wmma<<<dim3(D_MODEL / 64, ROWS / 16), 128, 0, stream>>>(
      attn, o_w, o_b, out, D_MODEL, D_MODEL);
}